// Att_2_layer1_11879879544270
// MI455X (gfx1250) — compile-verified
//
#include <hip/hip_runtime.h>

// Problem constants (from reference)
#define BB    128
#define SS    4
#define TT    16
#define KBOX  36
#define VD    256
#define QD    256
#define HH    256
#define NB    (BB * SS * TT)     // 8192
#define NBS   (BB * SS)          // 512
#define KPAD  48                 // 36 rounded up to 3 WMMA tiles
#define LSTR  260                // LDS row stride (floats) to avoid bank conflicts

typedef __attribute__((ext_vector_type(2))) float v2f;
typedef __attribute__((ext_vector_type(8))) float v8f;
typedef int v4i_ __attribute__((vector_size(16)));

// ---------------------------------------------------------------------------
// Async global->LDS staging (CDNA5 GLOBAL_LOAD_ASYNC_TO_LDS_B128, ASYNCcnt)
// with a safe fallback to VGPR-bounce copies if the builtin is unavailable.
// Builtin signature (from hipcc diagnostic): param0 = v4i addrspace(1)*.
// ---------------------------------------------------------------------------
#if defined(__has_builtin)
#if __has_builtin(__builtin_amdgcn_global_load_async_to_lds_b128)
#define HAVE_ASYNC_LDS 1
#endif
#endif

__device__ __forceinline__ void stage_b128(void* lds_dst, const void* gsrc) {
#ifdef HAVE_ASYNC_LDS
  __builtin_amdgcn_global_load_async_to_lds_b128(
      (__attribute__((address_space(1))) v4i_*)gsrc,
      (__attribute__((address_space(3))) v4i_*)lds_dst, 0, 0);
#else
  *(float4*)lds_dst = *(const float4*)gsrc;
#endif
}

__device__ __forceinline__ void stage_wait() {
#ifdef HAVE_ASYNC_LDS
#if defined(__has_builtin) && __has_builtin(__builtin_amdgcn_s_wait_asynccnt)
  __builtin_amdgcn_s_wait_asynccnt(0);
#else
  asm volatile("s_wait_asynccnt 0x0" ::: "memory");
#endif
#endif
}

// ---------------------------------------------------------------------------
// Kernel 0: lengths + exclusive prefix over the 512 (b,s) pairs
// ---------------------------------------------------------------------------
__global__ __launch_bounds__(512) void scan_kernel(const int* __restrict__ tags,
                                                   int* __restrict__ offs,
                                                   int* __restrict__ lens) {
  __shared__ int sdat[NBS];
  const int tid = threadIdx.x;
  int sum = 0;
#pragma unroll
  for (int t = 0; t < TT; ++t) sum += tags[tid * TT + t];
  lens[tid] = sum;
  sdat[tid] = sum;
  __syncthreads();
  for (int off = 1; off < NBS; off <<= 1) {
    int prev = (tid >= off) ? sdat[tid - off] : 0;
    __syncthreads();
    sdat[tid] += prev;
    __syncthreads();
  }
  offs[tid] = sdat[tid] - sum;  // exclusive scan
}

// ---------------------------------------------------------------------------
// Kernel 1: sq[n,h] = relu(q[n,:] . Wq[h,:] + bq[h]) * Wl[h]
// WMMA f32 16x16x4: A = 16 q-rows (LDS), B = Wq^T (global)
// ---------------------------------------------------------------------------
__global__ __launch_bounds__(256) void sq_kernel(const float* __restrict__ q,
                                                 const float* __restrict__ Wq,
                                                 const float* __restrict__ bq,
                                                 const float* __restrict__ Wl,
                                                 float* __restrict__ sq) {
  __shared__ float q_lds[16 * LSTR];
  const int tid = threadIdx.x;
  const int n0  = blockIdx.x * 16;

  // stage 16 x 256 q rows
  for (int i = tid; i < 16 * 64; i += 256) {
    const int r  = i >> 6;
    const int c4 = (i & 63) << 2;
    stage_b128(&q_lds[r * LSTR + c4], q + (size_t)(n0 + r) * QD + c4);
  }
  stage_wait();
  __syncthreads();

  const int lane   = tid & 31;
  const int wave   = tid >> 5;
  const int lane16 = lane & 15;
  const int half   = lane >> 4;

  for (int ht = wave * 2; ht < wave * 2 + 2; ++ht) {
    const int h = ht * 16 + lane16;                 // this lane's h column
    const float* wrow = Wq + (size_t)h * QD;
    v8f acc = {};
#pragma unroll 4
    for (int d0 = 0; d0 < QD; d0 += 4) {
      const int dk = d0 + 2 * half;
      v2f a = *(const v2f*)&q_lds[lane16 * LSTR + dk];  // A[m=lane16][k=dk..dk+1]
      v2f b = *(const v2f*)(wrow + dk);                 // B[k][n=h] = Wq[h][k]
      acc = __builtin_amdgcn_wmma_f32_16x16x4_f32(false, a, false, b, (short)0, acc,
                                                  false, false);
    }
    const float bqh = bq[h];
    const float wlh = Wl[h];
#pragma unroll
    for (int r = 0; r < 8; ++r) {
      const int n = n0 + r + 8 * half;              // D row M = r + 8*half
      float val = acc[r] + bqh;
      val = val > 0.0f ? val : 0.0f;
      sq[(size_t)n * HH + h] = val * wlh;
    }
  }
}

// ---------------------------------------------------------------------------
// Kernel 2: per output row (b,s,t): gather src box-set, fused
// logits GEMM (transposed: D[h,m], A=Wv, B=v) + relu + scale + h-reduction,
// then masked softmax + renormalization over K=36.
// ---------------------------------------------------------------------------
__global__ __launch_bounds__(256) void logits_softmax_kernel(
    const float* __restrict__ v, const float* __restrict__ box_mask,
    const float* __restrict__ Wv, const float* __restrict__ bv,
    const float* __restrict__ bl, const float* __restrict__ sq,
    const int* __restrict__ offs, const int* __restrict__ lens,
    float* __restrict__ out) {
  __shared__ float v_lds[KPAD * LSTR];
  __shared__ float bv_lds[HH];
  __shared__ float sq_lds[HH];
  __shared__ float logits_lds[KPAD];
  __shared__ float sm_e[KBOX];
  __shared__ float sm_m[KBOX];
  __shared__ float sred[3];

  const int row = blockIdx.x;           // flat (b,s,t)
  const int b   = row / (SS * TT);
  const int bs  = row / TT;
  const int t   = row - bs * TT;
  const int tid = threadIdx.x;

  const int  len   = lens[bs];
  const bool valid = t < len;
  int src = offs[bs] + t;
  src = src < 0 ? 0 : (src > NB - 1 ? NB - 1 : src);

  if (tid < KPAD) logits_lds[tid] = 0.0f;
  if (tid < 64) {
    stage_b128(&bv_lds[tid * 4], bv + tid * 4);
    stage_b128(&sq_lds[tid * 4], sq + (size_t)src * HH + tid * 4);
  }
  // stage v[src]: 36 rows x 256 cols (async direct-to-LDS)
  for (int i = tid; i < KBOX * 64; i += 256) {
    const int r  = i >> 6;
    const int c4 = (i & 63) << 2;
    stage_b128(&v_lds[r * LSTR + c4], v + ((size_t)src * KBOX + r) * VD + c4);
  }
  // zero the 12 pad rows
  for (int i = tid; i < (KPAD - KBOX) * 64; i += 256) {
    const int r  = KBOX + (i >> 6);
    const int c4 = (i & 63) << 2;
    *(float4*)&v_lds[r * LSTR + c4] = make_float4(0.f, 0.f, 0.f, 0.f);
  }
  stage_wait();
  __syncthreads();

  if (valid) {
    const int lane   = tid & 31;
    const int wave   = tid >> 5;
    const int lane16 = lane & 15;
    const int half   = lane >> 4;

    float part0 = 0.0f, part1 = 0.0f, part2 = 0.0f;
    for (int ht = wave; ht < 16; ht += 8) {        // 2 h-tiles per wave
      const int hb = ht * 16;
      const float* arow = Wv + (size_t)(hb + lane16) * VD;  // A row = h
      v8f acc0 = {}, acc1 = {}, acc2 = {};
#pragma unroll 8
      for (int d0 = 0; d0 < VD; d0 += 4) {
        const int dk = d0 + 2 * half;
        v2f a  = *(const v2f*)(arow + dk);                       // A[h][d]
        v2f b0 = *(const v2f*)&v_lds[(0  + lane16) * LSTR + dk]; // B[d][m]
        v2f b1 = *(const v2f*)&v_lds[(16 + lane16) * LSTR + dk];
        v2f b2 = *(const v2f*)&v_lds[(32 + lane16) * LSTR + dk];
        acc0 = __builtin_amdgcn_wmma_f32_16x16x4_f32(false, a, false, b0, (short)0,
                                                     acc0, false, false);
        acc1 = __builtin_amdgcn_wmma_f32_16x16x4_f32(false, a, false, b1, (short)0,
                                                     acc1, false, false);
        acc2 = __builtin_amdgcn_wmma_f32_16x16x4_f32(false, a, false, b2, (short)0,
                                                     acc2, false, false);
      }
      // relu + bias + scale by sq, reduce over the 8 h's held in this lane
#pragma unroll
      for (int r = 0; r < 8; ++r) {
        const int h = hb + r + 8 * half;           // D row M = h within tile
        const float bvh = bv_lds[h];
        const float sqh = sq_lds[h];
        float c0 = acc0[r] + bvh; c0 = c0 > 0.f ? c0 : 0.f; part0 += c0 * sqh;
        float c1 = acc1[r] + bvh; c1 = c1 > 0.f ? c1 : 0.f; part1 += c1 * sqh;
        float c2 = acc2[r] + bvh; c2 = c2 > 0.f ? c2 : 0.f; part2 += c2 * sqh;
      }
    }
    // fold the two half-wave h-groups together (same m column in both halves)
    part0 += __shfl_xor(part0, 16, 32);
    part1 += __shfl_xor(part1, 16, 32);
    part2 += __shfl_xor(part2, 16, 32);
    if (lane < 16) {
      atomicAdd(&logits_lds[lane16],      part0);
      atomicAdd(&logits_lds[16 + lane16], part1);
      atomicAdd(&logits_lds[32 + lane16], part2);
    }
  }
  __syncthreads();

  // masked softmax + renorm over K=36
  const float bl0 = bl[0];
  if (tid < KBOX) {
    const float lg = valid ? (logits_lds[tid] + bl0) : 0.0f;
    const float mk = box_mask[b * KBOX + tid];
    sm_m[tid] = mk;
    sm_e[tid] = lg * mk;
  }
  __syncthreads();
  if (tid == 0) {
    float mx = sm_e[0];
    for (int k = 1; k < KBOX; ++k) mx = fmaxf(mx, sm_e[k]);
    sred[0] = mx;
  }
  __syncthreads();
  if (tid < KBOX) sm_e[tid] = __expf(sm_e[tid] - sred[0]);
  __syncthreads();
  if (tid == 0) {
    float sa = 0.0f, smk = 0.0f;
    for (int k = 0; k < KBOX; ++k) {
      sa  += sm_e[k];
      smk += sm_e[k] * sm_m[k];
    }
    sred[1] = sa;
    sred[2] = smk;
  }
  __syncthreads();
  if (tid < KBOX) {
    const float sa    = sred[1];
    const float w1    = sm_e[tid] * sm_m[tid] / sa;
    const float sumw1 = sred[2] / sa;
    out[(size_t)row * KBOX + tid] = w1 / (sumw1 + 1e-13f);
  }
}

// ---------------------------------------------------------------------------
extern "C" void kernel_launch(void* const* d_in, const int* in_sizes, int n_in,
                              void* d_out, int out_size, void* d_ws, size_t ws_size,
                              hipStream_t stream) {
  const float* v        = (const float*)d_in[0];
  const float* q        = (const float*)d_in[1];
  const float* box_mask = (const float*)d_in[2];
  const int*   tags     = (const int*)d_in[3];
  const float* Wv       = (const float*)d_in[4];
  const float* bv       = (const float*)d_in[5];
  const float* Wq       = (const float*)d_in[6];
  const float* bq       = (const float*)d_in[7];
  const float* Wl       = (const float*)d_in[8];
  const float* bl       = (const float*)d_in[9];
  float* out = (float*)d_out;

  // workspace layout: sq (NB*H floats) | offs (512 ints) | lens (512 ints)
  float* sq   = (float*)d_ws;
  int*   offs = (int*)((char*)d_ws + (size_t)NB * HH * sizeof(float));
  int*   lens = offs + NBS;

  scan_kernel<<<1, NBS, 0, stream>>>(tags, offs, lens);
  sq_kernel<<<NB / 16, 256, 0, stream>>>(q, Wq, bq, Wl, sq);
  logits_softmax_kernel<<<NB, 256, 0, stream>>>(v, box_mask, Wv, bv, bl, sq, offs,
                                                lens, out);
}